// KeypointLoss_6030134084337
// MI455X (gfx1250) — compile-verified
//
#include <hip/hip_runtime.h>
#include <hip/hip_bf16.h>
#include <cstdint>

// Problem constants (from reference setup_inputs)
#define BB 32
#define SS 4
#define KK 11
#define CC 9
#define HH 128
#define WW 128
#define HW (HH * WW)          // 16384
#define HALF (HW / 2)         // 8192 floats = 32KB per TDM chunk
#define NBSK (BB * SS * KK)   // 1408
#define NBS  (BB * SS)        // 128

typedef __attribute__((ext_vector_type(2))) float v2f;
typedef __attribute__((ext_vector_type(4))) float v4f;
typedef __attribute__((ext_vector_type(8))) float v8f;
typedef __attribute__((ext_vector_type(4))) unsigned int v4u;
typedef __attribute__((ext_vector_type(4))) int v4i;
typedef __attribute__((ext_vector_type(8))) int v8i;

// Issue a TDM 1-D (1-row 2-D) tile load: nelem f32 from gptr to the LDS byte address in lds_off.
// D# packing per CDNA5 ISA ch.8: group0 = {count,lds_addr,global_addr,type=2},
// group1 = {data_size=4B, tensor_dim0=nelem, tensor_dim1=1, tile_dim0=nelem,
//           tile_dim1=1, tensor_dim0_stride=nelem}. Groups 2/3 unused (<=2D).
// This toolchain exposes the 6-arg builtin: (g0, g1, g2, g3, g_extra, cpol).
__device__ __forceinline__ void tdm_load_1d(unsigned lds_off, const void* gptr,
                                            unsigned nelem) {
  unsigned long long ga = (unsigned long long)(uintptr_t)gptr;
  v4u g0;
  g0.x = 1u;                                            // count=1, user mode
  g0.y = lds_off;                                       // LDS byte address
  g0.z = (unsigned)ga;                                  // global_addr[31:0]
  g0.w = (unsigned)((ga >> 32) & 0x1FFFFFFull) | 0x80000000u; // [56:32] | type=2
  v8i g1;
  g1[0] = (int)(2u << 16);                              // data_size=2 (4 bytes)
  g1[1] = (int)((nelem & 0xFFFFu) << 16);               // tensor_dim0[15:0]
  g1[2] = (int)((nelem >> 16) | (1u << 16));            // dim0[31:16] | dim1=1
  g1[3] = (int)((nelem & 0xFFFFu) << 16);               // tile_dim0
  g1[4] = 1;                                            // tile_dim1=1, tile_dim2=0
  g1[5] = (int)nelem;                                   // tensor_dim0_stride[31:0]
  g1[6] = 0;
  g1[7] = 0;
  v4i gz4 = {0, 0, 0, 0};
  v8i gz8 = {0, 0, 0, 0, 0, 0, 0, 0};
  __builtin_amdgcn_tensor_load_to_lds(g0, g1, gz4, gz4, gz8, 0);
}

// Kernel A: one workgroup per (b,s,k). Fused sum((pred-gt)^2) + argmax(pred).
// gt tile staged global->LDS by the Tensor Data Mover; pred streamed with NT
// B128 loads. Per-wave sumsq partials folded with one V_WMMA_F32_16X16X4_F32
// (A = 64 lane partials, B = ones -> row sums): exact f32 reduction.
__global__ __launch_bounds__(256) void keyloss_hm_stage(
    const float* __restrict__ hm_preds,   // [B,S,K,H,W]
    const float* __restrict__ gts,        // [B,K,H,W]
    float* __restrict__ ws_sumsq,         // [B*S*K]
    float* __restrict__ ws_maxval,        // [B*S*K]
    int*   __restrict__ ws_maxidx)        // [B*S*K]
{
  __shared__ float gt_tile[HALF];
  __shared__ float wsum_lds[8];
  __shared__ float wmax_lds[8];
  __shared__ int   widx_lds[8];

  const int bsk = blockIdx.x;
  const int b   = bsk / (SS * KK);
  const int k   = bsk % KK;
  const int t   = threadIdx.x;

  const float* pred = hm_preds + (size_t)bsk * HW;
  const float* gt   = gts + ((size_t)b * KK + k) * HW;
  const v4f* pred4  = (const v4f*)pred;

  float acc0 = 0.f, acc1 = 0.f;
  float bestv = -3.402823466e+38f;
  int   besti = 0;

  for (int h = 0; h < 2; ++h) {
    if (t == 0) {
      tdm_load_1d((unsigned)(uintptr_t)&gt_tile[0], gt + h * HALF, HALF);
      __builtin_amdgcn_s_wait_tensorcnt(0);
    }
    __syncthreads();                       // gt chunk visible to all waves
#pragma unroll
    for (int it = 0; it < 8; ++it) {
      const int f = it * 256 + t;          // float4 index within chunk [0,2048)
      v4f p = __builtin_nontemporal_load(&pred4[h * (HALF / 4) + f]);
      v4f g = *(const v4f*)&gt_tile[f * 4];
      v4f d = p - g;
      const float ss = d.x * d.x + d.y * d.y + d.z * d.z + d.w * d.w;
      if (it & 1) acc1 += ss; else acc0 += ss;
      const int base = h * HALF + f * 4;   // strictly increasing per thread ->
      if (p.x > bestv) { bestv = p.x; besti = base + 0; }  // first-max kept
      if (p.y > bestv) { bestv = p.y; besti = base + 1; }
      if (p.z > bestv) { bestv = p.z; besti = base + 2; }
      if (p.w > bestv) { bestv = p.w; besti = base + 3; }
    }
    __syncthreads();                       // done reading before next TDM write
  }

  // Wave-level sumsq reduction on the matrix pipe: D = A(16x4 partials) x ones.
  v2f a = {acc0, acc1};
  v2f ones = {1.0f, 1.0f};
  v8f c = {0.f, 0.f, 0.f, 0.f, 0.f, 0.f, 0.f, 0.f};
  v8f dm = __builtin_amdgcn_wmma_f32_16x16x4_f32(
      false, a, false, ones, (short)0, c, false, false);
  float wsum = dm[0] + dm[1] + dm[2] + dm[3] + dm[4] + dm[5] + dm[6] + dm[7];
  wsum += __shfl_xor(wsum, 16, 32);        // lanes<16 hold M0-7, >=16 hold M8-15

  // Wave-level argmax with jnp.argmax tie-break (lowest flat index wins).
  for (int off = 16; off > 0; off >>= 1) {
    float ov = __shfl_down(bestv, off, 32);
    int   oi = __shfl_down(besti, off, 32);
    if (ov > bestv || (ov == bestv && oi < besti)) { bestv = ov; besti = oi; }
  }

  const int wid = t >> 5, lane = t & 31;
  if (lane == 0) { wsum_lds[wid] = wsum; wmax_lds[wid] = bestv; widx_lds[wid] = besti; }
  __syncthreads();
  if (t == 0) {
    float tot = 0.f;
    float bv = wmax_lds[0]; int bi = widx_lds[0];
#pragma unroll
    for (int w = 0; w < 8; ++w) {
      tot += wsum_lds[w];
      if (wmax_lds[w] > bv || (wmax_lds[w] == bv && widx_lds[w] < bi)) {
        bv = wmax_lds[w]; bi = widx_lds[w];
      }
    }
    ws_sumsq[bsk]  = tot;
    ws_maxval[bsk] = bv;
    ws_maxidx[bsk] = bi;
  }
}

// Kernel B: one wave per (b,s). Lane k<11 computes that keypoint's label-loss
// terms and its hm partial; wave-reduces both; lane 0 writes final outputs.
__global__ __launch_bounds__(32) void keyloss_lb_stage(
    const float* __restrict__ lb_preds,   // [B,S,9,H,W]
    const float* __restrict__ hm_preds,   // [B,S,K,H,W]
    const float* __restrict__ labels,     // [B,K,11]
    const float* __restrict__ ws_sumsq,
    const int*   __restrict__ ws_maxidx,
    float* __restrict__ out)              // [0,128)=hm_loss, [128,256)=lb_loss
{
  const int bs   = blockIdx.x;            // b*S + s
  const int b    = bs / SS;
  const int lane = threadIdx.x;

  float hm = 0.f, lb = 0.f;
  if (lane < KK) {
    const int bsk = bs * KK + lane;
    hm = ws_sumsq[bsk];
    const int idx = ws_maxidx[bsk];
    const int x = idx / HH;               // reference: idx // m, m = H
    const int y = idx % HH;
    const int pos = x * WW + y;           // == idx when H == W
    const float* lab = labels + ((size_t)b * KK + lane) * 11;
    const float gx = lab[9], gy = lab[10];
    const bool valid =
        (gx >= 0.f) && (gy >= 0.f) && (gx < (float)HH) && (gy < (float)WW);
    const float conf = hm_preds[(size_t)bsk * HW + pos];
    const float dx = gx - (float)x, dy = gy - (float)y;
    const float xyl = dx * dx + dy * dy;
    const float cl  = (1.f - conf) * (1.f - conf);
    float cls = 0.f;
    const float* pb = lb_preds + (size_t)bs * CC * HW + pos;
#pragma unroll
    for (int c = 0; c < CC; ++c) {
      const float d = pb[(size_t)c * HW] - lab[c];
      cls += d * d;
    }
    lb = valid ? (cls + xyl + cl) : 0.f;
  }
  for (int off = 16; off > 0; off >>= 1) {
    hm += __shfl_down(hm, off, 32);
    lb += __shfl_down(lb, off, 32);
  }
  if (lane == 0) {
    out[bs]       = hm;                   // hm_loss [B,S]
    out[NBS + bs] = lb;                   // lb_loss [B,S]
  }
}

extern "C" void kernel_launch(void* const* d_in, const int* in_sizes, int n_in,
                              void* d_out, int out_size, void* d_ws, size_t ws_size,
                              hipStream_t stream) {
  const float* hm_preds = (const float*)d_in[0]; // [32,4,11,128,128]
  const float* lb_preds = (const float*)d_in[1]; // [32,4,9,128,128]
  const float* gts      = (const float*)d_in[2]; // [32,11,128,128]
  const float* labels   = (const float*)d_in[3]; // [32,11,11]
  float* out = (float*)d_out;                    // 256 floats: hm | lb

  float* ws_sumsq  = (float*)d_ws;
  float* ws_maxval = ws_sumsq + NBSK;
  int*   ws_maxidx = (int*)(ws_maxval + NBSK);

  keyloss_hm_stage<<<NBSK, 256, 0, stream>>>(hm_preds, gts, ws_sumsq,
                                             ws_maxval, ws_maxidx);
  keyloss_lb_stage<<<NBS, 32, 0, stream>>>(lb_preds, hm_preds, labels,
                                           ws_sumsq, ws_maxidx, out);
}